// MACELayer_17935783428301
// MI455X (gfx1250) — compile-verified
//
#include <hip/hip_runtime.h>
#include <math.h>

typedef __attribute__((ext_vector_type(2))) float v2f;
typedef __attribute__((ext_vector_type(8))) float v8f;

// D = A(16x4 f32) * B(4x16 f32) + C(16x16 f32), wave32.
// A per-lane: a.x=A[m=lane&15][kbase+2*(lane>>4)], a.y = next k
// B per-lane: b.x=B[kbase+2*(lane>>4)][n=lane&15], b.y = next k
// D slot r  : element (M = 8*(lane>>4)+r, N = lane&15)
__device__ __forceinline__ v8f wmma4(v2f a, v2f b, v8f c) {
    return __builtin_amdgcn_wmma_f32_16x16x4_f32(
        false, a, false, b, (short)0, c, false, false);
}

#define FDIM 64

// ---------------------------------------------------------------------------
// Kernel A: node up-projection.  s_up = nfs @ Wu_s ;  v_up[:,:,c] = nfv[:,:,c] @ Wu_v
// block = 32 (1 wave), grid = (ceil(N/16), 4).  blockIdx.y = component
// (0=s, 1..3=v_xyz) -> scalar-uniform branch, no exec divergence in the
// WMMA loops.  Each wave: 4 N-tiles x 16 K-steps WMMA.
// ---------------------------------------------------------------------------
__global__ void k_node_up(const float* __restrict__ nfs,
                          const float* __restrict__ nfv,
                          const float* __restrict__ Wus,
                          const float* __restrict__ Wuv,
                          float* __restrict__ s_up,
                          float* __restrict__ v_up,
                          int N) {
    const int lane = threadIdx.x & 31;
    const int half = lane >> 4;
    const int col  = lane & 15;
    const int n0   = blockIdx.x * 16;
    const int comp = blockIdx.y;               // SGPR -> scalar branch

    int row = n0 + col; if (row >= N) row = N - 1;   // clamp, guard on store

    if (comp == 0) {
        const float* __restrict__ A = nfs + (long)row * FDIM;
        #pragma unroll
        for (int t = 0; t < 4; ++t) {
            v8f acc = {};
            const int n = t * 16 + col;
            #pragma unroll
            for (int kk = 0; kk < 16; ++kk) {
                const int k = kk * 4 + half * 2;
                v2f a; a.x = A[k]; a.y = A[k + 1];
                v2f b; b.x = Wus[k * FDIM + n]; b.y = Wus[(k + 1) * FDIM + n];
                acc = wmma4(a, b, acc);
            }
            #pragma unroll
            for (int r = 0; r < 8; ++r) {
                const int node = n0 + half * 8 + r;
                if (node < N) s_up[(long)node * FDIM + n] = acc[r];
            }
        }
    } else {
        const int c = comp - 1;
        const float* __restrict__ A = nfv + (long)row * FDIM * 3 + c;
        #pragma unroll
        for (int t = 0; t < 4; ++t) {
            v8f acc = {};
            const int n = t * 16 + col;
            #pragma unroll
            for (int kk = 0; kk < 16; ++kk) {
                const int k = kk * 4 + half * 2;
                v2f a; a.x = A[k * 3]; a.y = A[k * 3 + 3];
                v2f b; b.x = Wuv[k * FDIM + n]; b.y = Wuv[(k + 1) * FDIM + n];
                acc = wmma4(a, b, acc);
            }
            #pragma unroll
            for (int r = 0; r < 8; ++r) {
                const int node = n0 + half * 8 + r;
                if (node < N) v_up[((long)node * FDIM + n) * 3 + c] = acc[r];
            }
        }
    }
}

// ---------------------------------------------------------------------------
// Kernel B: fused edge kernel.  Per wave: 16 edges.
//   GEMM1: h = silu(rE[16x8] @ W_r1[8x64])           (8 WMMA)
//   GEMM2: w = h[16x64] @ W_r2[64x320]               (320 WMMA, 4 f-tiles x 5 paths)
//   gather xs/xv (L2-resident), CG tensor product, atomic scatter to agg.
// block = 256 (8 waves), grid.x = ceil(E/128).
// ---------------------------------------------------------------------------
__global__ void k_edge(const float* __restrict__ vectors,
                       const float* __restrict__ rE,
                       const float* __restrict__ Wr1,
                       const float* __restrict__ Wr2,
                       const float* __restrict__ s_up,
                       const float* __restrict__ v_up,
                       const int* __restrict__ senders,
                       const int* __restrict__ receivers,
                       float* __restrict__ agg_s,
                       float* __restrict__ agg_v,
                       int E) {
    __shared__ float hbuf[8][16][FDIM];   // 32 KB: silu(h), A-layout staging
    __shared__ float ybuf[8][16][3];
    __shared__ int   sbuf[8][16];
    __shared__ int   rbuf[8][16];

    const int wave = threadIdx.x >> 5;
    const int lane = threadIdx.x & 31;
    const int half = lane >> 4;
    const int col  = lane & 15;
    const long e0  = ((long)blockIdx.x * 8 + wave) * 16;

    if (lane < 16) {                       // edge metadata (reconverges)
        long e = e0 + lane; if (e >= E) e = E - 1;
        const float vx = vectors[e * 3 + 0];
        const float vy = vectors[e * 3 + 1];
        const float vz = vectors[e * 3 + 2];
        const float nrm = sqrtf(vx * vx + vy * vy + vz * vz) + 1e-9f;
        ybuf[wave][lane][0] = vx / nrm;
        ybuf[wave][lane][1] = vy / nrm;
        ybuf[wave][lane][2] = vz / nrm;
        sbuf[wave][lane] = senders[e];
        rbuf[wave][lane] = receivers[e];
    }

    long erow = e0 + col; if (erow >= E) erow = E - 1;

    // GEMM1 + silu -> LDS (A layout for GEMM2)
    #pragma unroll
    for (int t = 0; t < 4; ++t) {
        v8f acc = {};
        #pragma unroll
        for (int kk = 0; kk < 2; ++kk) {
            const int k = kk * 4 + half * 2;
            v2f a; a.x = rE[erow * 8 + k]; a.y = rE[erow * 8 + k + 1];
            v2f b; b.x = Wr1[k * FDIM + t * 16 + col];
                   b.y = Wr1[(k + 1) * FDIM + t * 16 + col];
            acc = wmma4(a, b, acc);
        }
        #pragma unroll
        for (int r = 0; r < 8; ++r) {
            const float x = acc[r];
            hbuf[wave][half * 8 + r][t * 16 + col] = x / (1.0f + __expf(-x));
        }
    }
    __syncthreads();

    // GEMM2 per f-tile: 5 path weight tiles, then message formation + scatter
    #pragma unroll 1
    for (int ft = 0; ft < 4; ++ft) {
        v8f wacc[5];
        #pragma unroll
        for (int p = 0; p < 5; ++p) {
            v8f acc = {};
            const int n = p * FDIM + ft * 16 + col;
            #pragma unroll
            for (int kk = 0; kk < 16; ++kk) {
                const int k = kk * 4 + half * 2;
                v2f a; a.x = hbuf[wave][col][k]; a.y = hbuf[wave][col][k + 1];
                v2f b; b.x = Wr2[k * 320 + n];   b.y = Wr2[(k + 1) * 320 + n];
                acc = wmma4(a, b, acc);
            }
            wacc[p] = acc;
        }
        #pragma unroll
        for (int r = 0; r < 8; ++r) {
            const int  m = half * 8 + r;
            const long e = e0 + m;
            if (e < E) {
                const int f   = ft * 16 + col;
                const int snd = sbuf[wave][m];
                const int rcv = rbuf[wave][m];
                const float yx = ybuf[wave][m][0];
                const float yy = ybuf[wave][m][1];
                const float yz = ybuf[wave][m][2];
                const float xs = s_up[(long)snd * FDIM + f];
                const float* vp = v_up + ((long)snd * FDIM + f) * 3;
                const float xv0 = vp[0], xv1 = vp[1], xv2 = vp[2];
                const float w0 = wacc[0][r], w1 = wacc[1][r], w2 = wacc[2][r];
                const float w3 = wacc[3][r], w4 = wacc[4][r];
                const float dot = xv0 * yx + xv1 * yy + xv2 * yz;
                const float ms  = w0 * xs + w3 * dot;
                const float cx = xv1 * yz - xv2 * yy;
                const float cy = xv2 * yx - xv0 * yz;
                const float cz = xv0 * yy - xv1 * yx;
                const float mv0 = w1 * xs * yx + w2 * xv0 + w4 * cx;
                const float mv1 = w1 * xs * yy + w2 * xv1 + w4 * cy;
                const float mv2 = w1 * xs * yz + w2 * xv2 + w4 * cz;
                atomicAdd(&agg_s[(long)rcv * FDIM + f], ms);
                float* ap = agg_v + ((long)rcv * FDIM + f) * 3;
                atomicAdd(ap + 0, mv0);
                atomicAdd(ap + 1, mv1);
                atomicAdd(ap + 2, mv2);
            }
        }
    }
}

// ---------------------------------------------------------------------------
// Kernel C: node post.  Per wave: 16 nodes.
//   stage1: (agg @ Wd) * inv   -> product basis (elementwise in D layout) -> LDS
//   stage2: (p @ WL) + species-indexed skip (VALU) -> outputs; then readout.
// block = 64 (2 waves), grid.x = ceil(N/32).  LDS: 2 * 5 * 4KB = 40 KB.
// ---------------------------------------------------------------------------
__global__ void k_node_post(const float* __restrict__ agg_s,
                            const float* __restrict__ agg_v,
                            const float* __restrict__ Wds,
                            const float* __restrict__ Wdv,
                            const float* __restrict__ wsym_s,
                            const float* __restrict__ wsym_v,
                            const float* __restrict__ WLs,
                            const float* __restrict__ WLv,
                            const float* __restrict__ Wsk_s,
                            const float* __restrict__ Wsk_v,
                            const float* __restrict__ Wout,
                            const float* __restrict__ nfs,
                            const float* __restrict__ nfv,
                            const int* __restrict__ specie,
                            float* __restrict__ out0,
                            float* __restrict__ out_fs,
                            float* __restrict__ out_fv,
                            int N) {
    __shared__ float psb[2][16][FDIM];
    __shared__ float pvb[2][3][16][FDIM];
    __shared__ float fsb[2][16][FDIM];

    const int wave = threadIdx.x >> 5;
    const int lane = threadIdx.x & 31;
    const int half = lane >> 4;
    const int col  = lane & 15;
    const int n0   = (blockIdx.x * 2 + wave) * 16;
    const float inv = 0.25f;   // 1/sqrt(16)

    int row = n0 + col; if (row >= N) row = N - 1;

    // ---- stage 1: linear-down + symmetric contraction (correlation 2)
    #pragma unroll 1
    for (int ct = 0; ct < 4; ++ct) {
        v8f ds = {}, dv0 = {}, dv1 = {}, dv2 = {};
        const int n = ct * 16 + col;
        #pragma unroll
        for (int kk = 0; kk < 16; ++kk) {
            const int k = kk * 4 + half * 2;
            v2f b;  b.x  = Wds[k * FDIM + n]; b.y  = Wds[(k + 1) * FDIM + n];
            v2f bv; bv.x = Wdv[k * FDIM + n]; bv.y = Wdv[(k + 1) * FDIM + n];
            v2f a;  const float* pa = agg_s + (long)row * FDIM + k;
            a.x = pa[0]; a.y = pa[1];
            ds = wmma4(a, b, ds);
            const float* pv = agg_v + ((long)row * FDIM + k) * 3;
            v2f a0; a0.x = pv[0]; a0.y = pv[3]; dv0 = wmma4(a0, bv, dv0);
            v2f a1; a1.x = pv[1]; a1.y = pv[4]; dv1 = wmma4(a1, bv, dv1);
            v2f a2; a2.x = pv[2]; a2.y = pv[5]; dv2 = wmma4(a2, bv, dv2);
        }
        #pragma unroll
        for (int r = 0; r < 8; ++r) {
            const int m = half * 8 + r;
            int node = n0 + m; if (node >= N) node = N - 1;
            const int sp = specie[node];
            const int f  = ct * 16 + col;
            const float sv = ds[r]  * inv;
            const float v0 = dv0[r] * inv;
            const float v1 = dv1[r] * inv;
            const float v2 = dv2[r] * inv;
            const float* wsp = wsym_s + (long)sp * 3 * FDIM + f;
            const float* wvp = wsym_v + (long)sp * 2 * FDIM + f;
            const float ws0 = wsp[0], ws1 = wsp[FDIM], ws2 = wsp[2 * FDIM];
            const float wv0 = wvp[0], wv1 = wvp[FDIM];
            psb[wave][m][f]    = ws0 * sv + ws1 * sv * sv
                               + ws2 * (v0 * v0 + v1 * v1 + v2 * v2);
            pvb[wave][0][m][f] = wv0 * v0 + wv1 * sv * v0;
            pvb[wave][1][m][f] = wv0 * v1 + wv1 * sv * v1;
            pvb[wave][2][m][f] = wv0 * v2 + wv1 * sv * v2;
        }
    }
    __syncthreads();

    // ---- stage 2: final linear + species skip (VALU) + store
    #pragma unroll 1
    for (int ct = 0; ct < 4; ++ct) {
        v8f fs = {}, fv0 = {}, fv1 = {}, fv2 = {};
        const int n = ct * 16 + col;
        #pragma unroll
        for (int kk = 0; kk < 16; ++kk) {
            const int k = kk * 4 + half * 2;
            v2f b;  b.x  = WLs[k * FDIM + n]; b.y  = WLs[(k + 1) * FDIM + n];
            v2f bv; bv.x = WLv[k * FDIM + n]; bv.y = WLv[(k + 1) * FDIM + n];
            v2f a;  a.x  = psb[wave][col][k];    a.y  = psb[wave][col][k + 1];
            fs = wmma4(a, b, fs);
            v2f a0; a0.x = pvb[wave][0][col][k]; a0.y = pvb[wave][0][col][k + 1];
            fv0 = wmma4(a0, bv, fv0);
            v2f a1; a1.x = pvb[wave][1][col][k]; a1.y = pvb[wave][1][col][k + 1];
            fv1 = wmma4(a1, bv, fv1);
            v2f a2; a2.x = pvb[wave][2][col][k]; a2.y = pvb[wave][2][col][k + 1];
            fv2 = wmma4(a2, bv, fv2);
        }
        #pragma unroll 1
        for (int r = 0; r < 8; ++r) {
            const int m = half * 8 + r;
            const int node = n0 + m;
            if (node < N) {
                const int sp = specie[node];
                float as = fs[r], a0 = fv0[r], a1 = fv1[r], a2 = fv2[r];
                const float* Ws = Wsk_s + (long)sp * FDIM * FDIM + n;
                const float* Wv = Wsk_v + (long)sp * FDIM * FDIM + n;
                const float* xs = nfs + (long)node * FDIM;
                const float* xv = nfv + (long)node * FDIM * 3;
                for (int f = 0; f < FDIM; ++f) {   // species-indexed skip_tp
                    const float wS = Ws[f * FDIM];
                    const float wV = Wv[f * FDIM];
                    as += xs[f] * wS;
                    a0 += xv[f * 3 + 0] * wV;
                    a1 += xv[f * 3 + 1] * wV;
                    a2 += xv[f * 3 + 2] * wV;
                }
                out_fs[(long)node * FDIM + n] = as;
                fsb[wave][m][n] = as;
                float* op = out_fv + ((long)node * FDIM + n) * 3;
                op[0] = a0; op[1] = a1; op[2] = a2;
            }
        }
    }
    __syncthreads();

    // ---- readout: out0[node] = fs . W_out
    if (lane < 16) {
        const int node = n0 + lane;
        if (node < N) {
            float acc = 0.0f;
            for (int g = 0; g < FDIM; ++g) acc += fsb[wave][lane][g] * Wout[g];
            out0[node] = acc;
        }
    }
}

// ---------------------------------------------------------------------------
extern "C" void kernel_launch(void* const* d_in, const int* in_sizes, int n_in,
                              void* d_out, int out_size, void* d_ws, size_t ws_size,
                              hipStream_t stream) {
    const float* vectors = (const float*)d_in[0];
    const float* nfs     = (const float*)d_in[1];
    const float* nfv     = (const float*)d_in[2];
    const float* rE      = (const float*)d_in[3];
    const float* Wus     = (const float*)d_in[4];
    const float* Wuv     = (const float*)d_in[5];
    const float* Wr1     = (const float*)d_in[6];
    const float* Wr2     = (const float*)d_in[7];
    const float* Wds     = (const float*)d_in[8];
    const float* Wdv     = (const float*)d_in[9];
    const float* wsym_s  = (const float*)d_in[10];
    const float* wsym_v  = (const float*)d_in[11];
    const float* WLs     = (const float*)d_in[12];
    const float* WLv     = (const float*)d_in[13];
    const float* Wsk_s   = (const float*)d_in[14];
    const float* Wsk_v   = (const float*)d_in[15];
    const float* Wout    = (const float*)d_in[16];
    const int*   specie  = (const int*)d_in[17];
    const int*   senders = (const int*)d_in[18];
    const int*   recvers = (const int*)d_in[19];

    const int N = in_sizes[1] / FDIM;       // 50000
    const int E = in_sizes[0] / 3;          // 800000

    float* ws    = (float*)d_ws;
    float* s_up  = ws;                                   // N*64
    float* v_up  = s_up + (size_t)N * FDIM;              // N*192
    float* agg_s = v_up + (size_t)N * FDIM * 3;          // N*64
    float* agg_v = agg_s + (size_t)N * FDIM;             // N*192

    float* out0   = (float*)d_out;                       // [N]
    float* out_fs = out0 + N;                            // [N,64]
    float* out_fv = out_fs + (size_t)N * FDIM;           // [N,64,3]

    // zero aggregation buffers (agg_s and agg_v are contiguous)
    hipMemsetAsync(agg_s, 0, (size_t)N * FDIM * 4 * sizeof(float), stream);

    k_node_up<<<dim3((N + 15) / 16, 4), dim3(32), 0, stream>>>(
        nfs, nfv, Wus, Wuv, s_up, v_up, N);

    k_edge<<<dim3((E + 127) / 128), dim3(256), 0, stream>>>(
        vectors, rE, Wr1, Wr2, s_up, v_up, senders, recvers, agg_s, agg_v, E);

    k_node_post<<<dim3((N + 31) / 32), dim3(64), 0, stream>>>(
        agg_s, agg_v, Wds, Wdv, wsym_s, wsym_v, WLs, WLv,
        Wsk_s, Wsk_v, Wout, nfs, nfv, specie, out0, out_fs, out_fv, N);
}